// SyncModule_3959959847448
// MI455X (gfx1250) — compile-verified
//
#include <hip/hip_runtime.h>
#include <math.h>

#define B_ 4
#define N_ 4096
#define C_ 64
#define R_ 16
#define JC 64

typedef float v2f __attribute__((ext_vector_type(2)));
typedef float v8f __attribute__((ext_vector_type(8)));

// ---------------------------------------------------------------------------
// Kernel 1: gate A = sigmoid(U @ U^T)   (N x N, shared over batch)
// ---------------------------------------------------------------------------
__global__ __launch_bounds__(256) void gate_kernel(const float* __restrict__ U,
                                                   float* __restrict__ A) {
    __shared__ float Ui[16][R_];
    __shared__ float Uj[16][R_];
    const int t  = threadIdx.x;
    const int il = t >> 4, jl = t & 15;
    const int i0 = blockIdx.y * 16, j0 = blockIdx.x * 16;
    Ui[il][jl] = U[(i0 + il) * R_ + jl];
    Uj[il][jl] = U[(j0 + il) * R_ + jl];
    __syncthreads();
    float dot = 0.f;
#pragma unroll
    for (int k = 0; k < R_; ++k) dot += Ui[il][k] * Uj[jl][k];
    A[(size_t)(i0 + il) * N_ + (j0 + jl)] = 1.0f / (1.0f + __expf(-dot));
}

// ---------------------------------------------------------------------------
// Kernel 2a: row term  deg[b,i] = sum_j W[b,i,j] * A[i,j]   (overwrites deg)
// ---------------------------------------------------------------------------
__global__ __launch_bounds__(256) void rowdeg_kernel(const float* __restrict__ W,
                                                     const float* __restrict__ A,
                                                     float* __restrict__ deg) {
    __shared__ float red[256];
    const int i = blockIdx.x, b = blockIdx.y, t = threadIdx.x;
    const float* wrow = W + ((size_t)b * N_ + i) * N_;
    const float* arow = A + (size_t)i * N_;
    float acc = 0.f;
    for (int j = t; j < N_; j += 256) acc += wrow[j] * arow[j];
    red[t] = acc;
    __syncthreads();
    for (int s = 128; s > 0; s >>= 1) {
        if (t < s) red[t] += red[t + s];
        __syncthreads();
    }
    if (t == 0) deg[(size_t)b * N_ + i] = red[0];
}

// ---------------------------------------------------------------------------
// Kernel 2b: column term  deg[b,i] += sum_j W[b,j,i] * A[j,i]  (atomic add)
// Fully coalesced: thread t owns column i0+t, sweeps rows j.
// ---------------------------------------------------------------------------
__global__ __launch_bounds__(256) void coldeg_kernel(const float* __restrict__ W,
                                                     const float* __restrict__ A,
                                                     float* __restrict__ deg) {
    const int t  = threadIdx.x;
    const int i  = blockIdx.x * 256 + t;
    const int b  = blockIdx.z;
    const int j0 = blockIdx.y * (N_ / 8);
    const float* wb = W + (size_t)b * N_ * N_;
    float acc = 0.f;
    for (int j = j0; j < j0 + N_ / 8; ++j)
        acc += wb[(size_t)j * N_ + i] * A[(size_t)j * N_ + i];
    atomicAdd(&deg[(size_t)b * N_ + i], acc);
}

// ---------------------------------------------------------------------------
// Kernel 2c: d = rsqrt(max(0.5*(row+col), eps))
// ---------------------------------------------------------------------------
__global__ __launch_bounds__(256) void finalize_d_kernel(const float* __restrict__ deg,
                                                         float* __restrict__ dvec) {
    const int idx = blockIdx.x * 256 + threadIdx.x;
    const float g = fmaxf(0.5f * deg[idx], 1e-12f);
    dvec[idx] = rsqrtf(g);
}

// ---------------------------------------------------------------------------
// Kernel 3: out[b,i,c] = w_scale * sum_j d_i d_j 0.5(W[i,j]+W[j,i]) A[i,j] x[b,j,c]
// 8 waves / block; block owns (b, 32 output rows); waves split {i-half, c-tile}.
// Inner product via V_WMMA_F32_16X16X4_F32.
// ---------------------------------------------------------------------------
__global__ __launch_bounds__(256) void couple_kernel(
    const float* __restrict__ W, const float* __restrict__ x,
    const float* __restrict__ A, const float* __restrict__ dvec,
    const float* __restrict__ wscale_p, float* __restrict__ out) {
    __shared__ float sWc[JC][33];  // W[b, j0+jl, i0+il]  (transposed tile, pad 33)
    __shared__ float sK[32][65];   // normalized symmetric K tile (pad 65)
    __shared__ float sX[JC][65];   // x[b, j0+jl, c]       (pad 65)

    const int t    = threadIdx.x;
    const int lane = t & 31;
    const int wv   = t >> 5;        // wave 0..7
    const int ih   = wv >> 2;       // i-half: 0 or 1
    const int c0   = (wv & 3) * 16; // c-tile base

    const int nblk = N_ / 32;
    const int b    = blockIdx.x / nblk;
    const int i0   = (blockIdx.x % nblk) * 32;

    const float wsc = wscale_p[0];
    const float* wb = W + (size_t)b * N_ * N_;
    const float* xb = x + (size_t)b * N_ * C_;
    const float* db = dvec + (size_t)b * N_;

    const int m  = lane & 15;       // WMMA A: row M / WMMA B,D: col N
    const int hi = lane >> 4;       // lane half
    const int kb = hi * 2;          // K-slice base within the 4-wide K step

    v8f acc = {0.f, 0.f, 0.f, 0.f, 0.f, 0.f, 0.f, 0.f};

    for (int j0 = 0; j0 < N_; j0 += JC) {
        __syncthreads();
        // stage W^T tile (coalesced 128B row segments) and x tile
#pragma unroll
        for (int k = 0; k < (JC * 32) / 256; ++k) {
            const int idx = t + k * 256;
            const int jl = idx >> 5, il = idx & 31;
            sWc[jl][il] = wb[(size_t)(j0 + jl) * N_ + (i0 + il)];
        }
#pragma unroll
        for (int k = 0; k < (JC * C_) / 256; ++k) {
            const int idx = t + k * 256;
            const int jl = idx >> 6, cc = idx & 63;
            sX[jl][cc] = xb[(size_t)(j0 + jl) * C_ + cc];
        }
        __syncthreads();
        // build normalized symmetric K tile (row-major W + A loads coalesced)
#pragma unroll
        for (int k = 0; k < (32 * JC) / 256; ++k) {
            const int idx = t + k * 256;
            const int il = idx >> 6, jl = idx & 63;
            const float wr = wb[(size_t)(i0 + il) * N_ + (j0 + jl)];
            const float ag = A[(size_t)(i0 + il) * N_ + (j0 + jl)];
            const float di = db[i0 + il];
            const float dj = db[j0 + jl];
            sK[il][jl] = 0.5f * wsc * di * dj * (wr + sWc[jl][il]) * ag;
        }
        __syncthreads();
        // 16 x (K=4) WMMA steps over this chunk
#pragma unroll 4
        for (int j4 = 0; j4 < JC; j4 += 4) {
            v2f afrag, bfrag;
            // A 16x4 layout: lane m = M; lanes<16 -> K=0,1 ; lanes>=16 -> K=2,3
            afrag.x = sK[ih * 16 + m][j4 + kb];
            afrag.y = sK[ih * 16 + m][j4 + kb + 1];
            // B 4x16 layout: VGPR0 -> K = kb, VGPR1 -> K = kb+1 ; lane m = N
            bfrag.x = sX[j4 + kb][c0 + m];
            bfrag.y = sX[j4 + kb + 1][c0 + m];
            acc = __builtin_amdgcn_wmma_f32_16x16x4_f32(
                false, afrag, false, bfrag, (short)0, acc, false, false);
        }
    }

    // D 16x16 layout: VGPR r -> M = r + hi*8 ; lane m = N
    float* ob = out + ((size_t)b * N_ + i0 + ih * 16) * C_;
#pragma unroll
    for (int r = 0; r < 8; ++r)
        ob[(size_t)(r + hi * 8) * C_ + c0 + m] = acc[r];
}

// ---------------------------------------------------------------------------
extern "C" void kernel_launch(void* const* d_in, const int* in_sizes, int n_in,
                              void* d_out, int out_size, void* d_ws, size_t ws_size,
                              hipStream_t stream) {
    const float* x      = (const float*)d_in[0];
    const float* W      = (const float*)d_in[1];
    const float* U      = (const float*)d_in[2];
    const float* wscale = (const float*)d_in[3];
    float* out = (float*)d_out;

    // workspace: A (N*N) | deg (B*N) | d (B*N)
    float* A   = (float*)d_ws;
    float* deg = A + (size_t)N_ * N_;
    float* dv  = deg + (size_t)B_ * N_;

    gate_kernel<<<dim3(N_ / 16, N_ / 16), 256, 0, stream>>>(U, A);
    rowdeg_kernel<<<dim3(N_, B_), 256, 0, stream>>>(W, A, deg);
    coldeg_kernel<<<dim3(N_ / 256, 8, B_), 256, 0, stream>>>(W, A, deg);
    finalize_d_kernel<<<(B_ * N_) / 256, 256, 0, stream>>>(deg, dv);
    couple_kernel<<<B_ * (N_ / 32), 256, 0, stream>>>(W, x, A, dv, wscale, out);
}